// SparseMHA_89163521065068
// MI455X (gfx1250) — compile-verified
//
#include <hip/hip_runtime.h>
#include <hip/hip_bf16.h>

#define HIDDEN 128
#define NHEADS 8
#define DHEAD 16

typedef __attribute__((ext_vector_type(2))) float v2f;
typedef __attribute__((ext_vector_type(8))) float v8f;

// ---------------------------------------------------------------------------
// Per-wave 16x16 f32 GEMM tile using V_WMMA_F32_16X16X4_F32 (wave32).
// Computes D[16x16] = A[16x128] * B[128x16] where B[k][n] = W[o0+n][k]
// (i.e. out = A @ W^T restricted to output columns [o0, o0+16)).
//
// ISA lane layouts (CDNA5 ISA):
//   A 16x4:  lanes 0-15 hold (A[M,k0+0], A[M,k0+1]) in 2 VGPRs,
//            lanes 16-31 hold (A[M,k0+2], A[M,k0+3])       -> contiguous b64
//   B 4x16:  mirrored with N in lanes                       -> contiguous b64
//   C/D:     VGPR r holds row M=r (lanes 0-15) / M=r+8 (lanes 16-31), col N=lane%16
// ---------------------------------------------------------------------------
__device__ __forceinline__ v8f wmma_tile_f32(const float* __restrict__ A,   // 16x128 row-major
                                             const float* __restrict__ W,   // 128x128 row-major
                                             int o0, int lane) {
  const int l16  = lane & 15;
  const int half = lane >> 4;
  const float* aBase = A + l16 * HIDDEN + half * 2;
  const float* bBase = W + (o0 + l16) * HIDDEN + half * 2;
  v8f acc = {0.f, 0.f, 0.f, 0.f, 0.f, 0.f, 0.f, 0.f};
#pragma unroll
  for (int k0 = 0; k0 < HIDDEN; k0 += 4) {
    v2f a = *(const v2f*)(aBase + k0);
    v2f b = *(const v2f*)(bBase + k0);
    acc = __builtin_amdgcn_wmma_f32_16x16x4_f32(false, a, false, b, (short)0, acc,
                                                false, false);
  }
  return acc;
}

// float atomic-max via order-preserving integer punning
__device__ __forceinline__ void atomicMaxF(float* addr, float val) {
  if (val >= 0.f)
    atomicMax((int*)addr, __float_as_int(val));
  else
    atomicMin((unsigned int*)addr, __float_as_uint(val));
}

// ---------------------------------------------------------------------------
// k0: initialize workspace: m = -inf, s = 0, accum = 0
// ---------------------------------------------------------------------------
__global__ void init_ws(unsigned int* __restrict__ mrow, float* __restrict__ srow,
                        float* __restrict__ accum, int n) {
  int i = blockIdx.x * blockDim.x + threadIdx.x;
  if (i < n * NHEADS) {
    mrow[i] = 0xFF800000u;  // -inf
    srow[i] = 0.f;
  }
  if (i < n * HIDDEN) accum[i] = 0.f;
}

// ---------------------------------------------------------------------------
// k1: fused Q/K/V projection.  grid = N/16 blocks, 256 threads (8 waves).
// Wave w computes the 16x16 output tile at columns [16w, 16w+16) for q, k, v.
// q additionally scaled by DHEAD^-0.5 = 0.25 (applied after bias, as in ref).
// ---------------------------------------------------------------------------
__global__ void qkv_proj(const float* __restrict__ h,
                         const float* __restrict__ Wq, const float* __restrict__ bq,
                         const float* __restrict__ Wk, const float* __restrict__ bk,
                         const float* __restrict__ Wv, const float* __restrict__ bv,
                         float* __restrict__ q, float* __restrict__ k,
                         float* __restrict__ v) {
  const int n0   = blockIdx.x * 16;
  const int wave = threadIdx.x >> 5;
  const int lane = threadIdx.x & 31;
  const int o0   = wave * 16;
  const int l16  = lane & 15;
  const int half = lane >> 4;
  const float* A = h + (size_t)n0 * HIDDEN;
  const int o    = o0 + l16;

#pragma unroll
  for (int m = 0; m < 3; ++m) {
    const float* W    = (m == 0) ? Wq : (m == 1) ? Wk : Wv;
    const float* bias = (m == 0) ? bq : (m == 1) ? bk : bv;
    float* out        = (m == 0) ? q : (m == 1) ? k : v;
    const float scale = (m == 0) ? 0.25f : 1.0f;

    v8f acc = wmma_tile_f32(A, W, o0, lane);
    const float bb = bias[o];
#pragma unroll
    for (int r = 0; r < 8; ++r)
      out[(size_t)(n0 + r + 8 * half) * HIDDEN + o] = (acc[r] + bb) * scale;
  }
}

// ---------------------------------------------------------------------------
// k2: SDDMM + running row max.  One thread per edge.
// Head layout: element (node, d, head) lives at column d*NHEADS + head, so
// column j belongs to head (j & 7).
// ---------------------------------------------------------------------------
__global__ void sddmm_max(const float* __restrict__ q, const float* __restrict__ k,
                          const int* __restrict__ rows, const int* __restrict__ cols,
                          float* __restrict__ scores, float* __restrict__ mrow, int E) {
  int e = blockIdx.x * blockDim.x + threadIdx.x;
  if (e >= E) return;
  int r = rows[e], c = cols[e];
  const float4* qa = (const float4*)(q + (size_t)r * HIDDEN);
  const float4* ka = (const float4*)(k + (size_t)c * HIDDEN);
  float acc[NHEADS];
#pragma unroll
  for (int i = 0; i < NHEADS; ++i) acc[i] = 0.f;
#pragma unroll
  for (int i = 0; i < HIDDEN / 4; ++i) {
    float4 qv = qa[i];
    float4 kv = ka[i];
    const int base = (i * 4) & 7;  // 0 or 4: heads base..base+3
    acc[base + 0] += qv.x * kv.x;
    acc[base + 1] += qv.y * kv.y;
    acc[base + 2] += qv.z * kv.z;
    acc[base + 3] += qv.w * kv.w;
  }
#pragma unroll
  for (int hh = 0; hh < NHEADS; ++hh) {
    scores[(size_t)e * NHEADS + hh] = acc[hh];
    atomicMaxF(&mrow[(size_t)r * NHEADS + hh], acc[hh]);
  }
}

// ---------------------------------------------------------------------------
// k3: e = exp(score - m[row]) in place + segment sum.  One thread per (edge,head).
// ---------------------------------------------------------------------------
__global__ void exp_sum(const int* __restrict__ rows, const float* __restrict__ mrow,
                        float* __restrict__ scores, float* __restrict__ srow, int EH) {
  int idx = blockIdx.x * blockDim.x + threadIdx.x;
  if (idx >= EH) return;
  int e  = idx >> 3;
  int hh = idx & 7;
  int r  = rows[e];
  float ex = __expf(scores[idx] - mrow[(size_t)r * NHEADS + hh]);
  scores[idx] = ex;
  atomicAdd(&srow[(size_t)r * NHEADS + hh], ex);
}

// ---------------------------------------------------------------------------
// k4: SpMM scatter.  One wave32 per edge; lane j covers columns 4j..4j+3.
// Lanes 0-7 compute the 8 normalized attention weights; everyone else gets
// them via wave shuffles.  4 global_atomic_add_f32 per lane.
// ---------------------------------------------------------------------------
__global__ void spmm_scatter(const int* __restrict__ rows, const int* __restrict__ cols,
                             const float* __restrict__ scores, const float* __restrict__ srow,
                             const float* __restrict__ v, float* __restrict__ accum, int E) {
  int gid  = blockIdx.x * blockDim.x + threadIdx.x;
  int e    = gid >> 5;
  int lane = threadIdx.x & 31;
  if (e >= E) return;
  int r = rows[e], c = cols[e];

  float attn = 0.f;
  if (lane < NHEADS)
    attn = scores[(size_t)e * NHEADS + lane] / srow[(size_t)r * NHEADS + lane];

  const int j0 = lane * 4;
  float4 vv = *(const float4*)(v + (size_t)c * HIDDEN + j0);
  float a0 = __shfl(attn, (j0 + 0) & 7, 32);
  float a1 = __shfl(attn, (j0 + 1) & 7, 32);
  float a2 = __shfl(attn, (j0 + 2) & 7, 32);
  float a3 = __shfl(attn, (j0 + 3) & 7, 32);

  float* dst = accum + (size_t)r * HIDDEN + j0;
  atomicAdd(dst + 0, a0 * vv.x);
  atomicAdd(dst + 1, a1 * vv.y);
  atomicAdd(dst + 2, a2 * vv.z);
  atomicAdd(dst + 3, a3 * vv.w);
}

// ---------------------------------------------------------------------------
// k5: output projection  out = accum @ Wo^T + bo  (same WMMA tile helper)
// ---------------------------------------------------------------------------
__global__ void out_proj(const float* __restrict__ accum,
                         const float* __restrict__ Wo, const float* __restrict__ bo,
                         float* __restrict__ out) {
  const int n0   = blockIdx.x * 16;
  const int wave = threadIdx.x >> 5;
  const int lane = threadIdx.x & 31;
  const int o0   = wave * 16;
  const int l16  = lane & 15;
  const int half = lane >> 4;

  v8f acc = wmma_tile_f32(accum + (size_t)n0 * HIDDEN, Wo, o0, lane);
  const int o    = o0 + l16;
  const float bb = bo[o];
#pragma unroll
  for (int r = 0; r < 8; ++r)
    out[(size_t)(n0 + r + 8 * half) * HIDDEN + o] = acc[r] + bb;
}

// ---------------------------------------------------------------------------
extern "C" void kernel_launch(void* const* d_in, const int* in_sizes, int n_in,
                              void* d_out, int out_size, void* d_ws, size_t ws_size,
                              hipStream_t stream) {
  const float* h  = (const float*)d_in[0];
  const int* rows = (const int*)d_in[1];
  const int* cols = (const int*)d_in[2];
  const float* Wq = (const float*)d_in[3];
  const float* bq = (const float*)d_in[4];
  const float* Wk = (const float*)d_in[5];
  const float* bk = (const float*)d_in[6];
  const float* Wv = (const float*)d_in[7];
  const float* bv = (const float*)d_in[8];
  const float* Wo = (const float*)d_in[9];
  const float* bo = (const float*)d_in[10];
  float* out = (float*)d_out;

  const int N = in_sizes[0] / HIDDEN;  // 50000
  const int E = in_sizes[1];           // 800000

  // Workspace layout (all fp32): q, k, v, scores[E,8], m[N,8], s[N,8], accum[N,128]
  float* ws     = (float*)d_ws;
  float* q      = ws; ws += (size_t)N * HIDDEN;
  float* k      = ws; ws += (size_t)N * HIDDEN;
  float* v      = ws; ws += (size_t)N * HIDDEN;
  float* scores = ws; ws += (size_t)E * NHEADS;
  float* mrow   = ws; ws += (size_t)N * NHEADS;
  float* srow   = ws; ws += (size_t)N * NHEADS;
  float* accum  = ws; ws += (size_t)N * HIDDEN;

  const dim3 blk(256);
  init_ws<<<(N * HIDDEN + 255) / 256, blk, 0, stream>>>((unsigned int*)mrow, srow, accum, N);
  qkv_proj<<<N / 16, blk, 0, stream>>>(h, Wq, bq, Wk, bk, Wv, bv, q, k, v);
  sddmm_max<<<(E + 255) / 256, blk, 0, stream>>>(q, k, rows, cols, scores, mrow, E);
  exp_sum<<<(E * NHEADS + 255) / 256, blk, 0, stream>>>(rows, mrow, scores, srow, E * NHEADS);
  spmm_scatter<<<(E * 32 + 255) / 256, blk, 0, stream>>>(rows, cols, scores, srow, v, accum, E);
  out_proj<<<N / 16, blk, 0, stream>>>(accum, Wo, bo, out);
}